// NewAttention_74182675136588
// MI455X (gfx1250) — compile-verified
//
#include <hip/hip_runtime.h>
#include <math.h>

// ---------------------------------------------------------------------------
// Problem constants (from reference): B=8, S=4096, E=512, H=8, d=64.
// Math: out = attn @ values @ (W_out @ W_in)^T, where attn = exp(-1)/Z[q]
// everywhere except a +-2 band around the diagonal (fp32-exact observation).
// ---------------------------------------------------------------------------
#define S_LEN 4096
#define BATCH 8
#define EMB   512

typedef float v2f __attribute__((ext_vector_type(2)));
typedef float v8f __attribute__((ext_vector_type(8)));

// ---------------------------------------------------------------------------
// WMMA f32 GEMM. Block = 256 threads (8 waves). Each block computes a
// [32 x N] output slab: A rows staged in LDS (stride 516 floats -> bank-
// conflict-free fragment reads for both 16-row halves); each wave owns
// 4 column tiles of 16 and both 16-row M-tiles (8 accumulators), so every
// B fragment fetched from L2 feeds two v_wmma_f32_16x16x4_f32 ops.
// TRANS_B=true : O = A[M,K] @ W^T, W stored [N,K]  (projection: values @ Wc^T)
// TRANS_B=false: O = A[M,K] @ W,   W stored [K,N]  (fold: W_out @ W_in)
// Requires K == 512, N == 512, M % 32 == 0.
// ---------------------------------------------------------------------------
template <bool TRANS_B>
__global__ __launch_bounds__(256) void gemm_wmma_f32(
    const float* __restrict__ A, const float* __restrict__ W,
    float* __restrict__ O, int M, int N, int K) {
  __shared__ float lds[32 * 516];  // 66,048 B (CDNA5: up to 320 KB per WG)
  const int tid = threadIdx.x;
  const int m0 = blockIdx.x * 32;
  const int ldsStride = 516;  // (516*row + k) % 64 == (4*row + k) % 64

  // Cooperative load of A slab [32 x K] into LDS (float4, fully coalesced).
  for (int f = tid * 4; f < 32 * K; f += 256 * 4) {
    const int r = f / K;
    const int c = f % K;
    float4 v = *(const float4*)(A + (size_t)(m0 + r) * K + c);
    *(float4*)&lds[r * ldsStride + c] = v;
  }
  __syncthreads();

  const int lane = tid & 31;
  const int wave = tid >> 5;
  const int half = lane >> 4;   // 0: K=k0..k0+1, 1: K=k0+2..k0+3 (A layout)
  const int l16  = lane & 15;

  v8f acc[2][4] = {{v8f{}, v8f{}, v8f{}, v8f{}}, {v8f{}, v8f{}, v8f{}, v8f{}}};

  for (int k0 = 0; k0 < K; k0 += 4) {
    const int kb = k0 + half * 2;
    // A fragments: lane<16 -> row l16, K=kb,kb+1 ; lane>=16 -> same row, K+2.
    v2f a0 = *(const v2f*)&lds[l16 * ldsStride + kb];
    v2f a1 = *(const v2f*)&lds[(16 + l16) * ldsStride + kb];

    v2f b[4];
#pragma unroll
    for (int t = 0; t < 4; ++t) {
      const int n = (wave * 4 + t) * 16 + l16;
      if (TRANS_B) {
        // B fragment element (k, n) = W[n, k]; contiguous float2 per lane.
        b[t] = *(const v2f*)(W + (size_t)n * K + kb);
      } else {
        b[t].x = W[(size_t)kb * N + n];
        b[t].y = W[(size_t)(kb + 1) * N + n];
      }
    }
#pragma unroll
    for (int t = 0; t < 4; ++t) {
      acc[0][t] = __builtin_amdgcn_wmma_f32_16x16x4_f32(
          false, a0, false, b[t], (short)0, acc[0][t], false, false);
      acc[1][t] = __builtin_amdgcn_wmma_f32_16x16x4_f32(
          false, a1, false, b[t], (short)0, acc[1][t], false, false);
    }
  }

  // C/D layout: VGPR j -> (M = mbase + j + 8*half, N = n).
#pragma unroll
  for (int mt = 0; mt < 2; ++mt) {
#pragma unroll
    for (int t = 0; t < 4; ++t) {
      const int n = (wave * 4 + t) * 16 + l16;
#pragma unroll
      for (int j = 0; j < 8; ++j) {
        const int m = m0 + mt * 16 + j + half * 8;
        O[(size_t)m * N + n] = acc[mt][t][j];
      }
    }
  }
}

// ---------------------------------------------------------------------------
// Deterministic two-phase column sum of P over the sequence axis.
// partial[b][seg][e] = sum over 128 rows; then reduce 32 segments.
// ---------------------------------------------------------------------------
__global__ __launch_bounds__(128) void colsum_partial_kernel(
    const float* __restrict__ P, float* __restrict__ partial) {
  const int seg = blockIdx.x;   // 0..31
  const int b   = blockIdx.y;   // 0..7
  const int e4  = threadIdx.x;  // 0..127
  const float* base = P + ((size_t)b * S_LEN + (size_t)seg * 128) * EMB;
  float4 s = {0.f, 0.f, 0.f, 0.f};
  for (int r = 0; r < 128; ++r) {
    float4 p = *(const float4*)(base + (size_t)r * EMB + e4 * 4);
    s.x += p.x; s.y += p.y; s.z += p.z; s.w += p.w;
  }
  *(float4*)(partial + ((size_t)b * 32 + seg) * EMB + e4 * 4) = s;
}

__global__ __launch_bounds__(128) void colsum_final_kernel(
    const float* __restrict__ partial, float* __restrict__ colsum) {
  const int b  = blockIdx.x;
  const int e4 = threadIdx.x;
  float4 s = {0.f, 0.f, 0.f, 0.f};
  for (int seg = 0; seg < 32; ++seg) {
    float4 p = *(const float4*)(partial + ((size_t)b * 32 + seg) * EMB + e4 * 4);
    s.x += p.x; s.y += p.y; s.z += p.z; s.w += p.w;
  }
  *(float4*)(colsum + (size_t)b * EMB + e4 * 4) = s;
}

// ---------------------------------------------------------------------------
// Banded softmax application:
// out[b,q,:] = ( e^-1 * colsum[b,:] + sum_{|d|<=2} (w(d)-e^-1) * P[b,q+d,:] ) / Z[q]
// with w(d) = expf(expf(-pi d^2) - 1) reproducing the reference fp32 chain
// (all |d|>=3 terms round to expf(-1) exactly in fp32).
// ---------------------------------------------------------------------------
__global__ __launch_bounds__(256) void band_apply_kernel(
    const float* __restrict__ P, const float* __restrict__ colsum,
    float* __restrict__ out) {
  const int E4 = EMB / 4;
  const size_t idx = (size_t)blockIdx.x * blockDim.x + threadIdx.x;
  const int e4 = (int)(idx % E4);
  const size_t bq = idx / E4;
  const int q = (int)(bq % S_LEN);
  const int b = (int)(bq / S_LEN);

  float w[3];
#pragma unroll
  for (int d = 0; d < 3; ++d) {
    const float l = expf(-0.5f * (float)(d * d) * 6.283185307179586f);
    w[d] = expf(l - 1.0f);
  }
  const float CF = expf(-1.0f);

  const int vlo = (q - 2 < 0) ? 0 : q - 2;
  const int vhi = (q + 2 > S_LEN - 1) ? S_LEN - 1 : q + 2;

  float Z = (float)(S_LEN - (vhi - vlo + 1)) * CF;

  float4 c = *(const float4*)(colsum + (size_t)b * EMB + e4 * 4);
  float4 acc;
  acc.x = c.x * CF; acc.y = c.y * CF; acc.z = c.z * CF; acc.w = c.w * CF;

  for (int v = vlo; v <= vhi; ++v) {
    int d = v - q; if (d < 0) d = -d;
    const float ww = w[d];
    Z += ww;
    const float dw = ww - CF;
    float4 p = *(const float4*)(P + ((size_t)b * S_LEN + v) * EMB + e4 * 4);
    acc.x += dw * p.x; acc.y += dw * p.y; acc.z += dw * p.z; acc.w += dw * p.w;
  }
  const float inv = 1.0f / Z;
  float4 o;
  o.x = acc.x * inv; o.y = acc.y * inv; o.z = acc.z * inv; o.w = acc.w * inv;
  *(float4*)(out + ((size_t)b * S_LEN + q) * EMB + e4 * 4) = o;
}

// ---------------------------------------------------------------------------
// Launch: fold -> projection -> colsum (2 phases) -> banded apply.
// ---------------------------------------------------------------------------
extern "C" void kernel_launch(void* const* d_in, const int* in_sizes, int n_in,
                              void* d_out, int out_size, void* d_ws, size_t ws_size,
                              hipStream_t stream) {
  const float* values = (const float*)d_in[0];
  // d_in[1] = keys, d_in[2] = queries: unused by the reference math.
  const float* W_in  = (const float*)d_in[3];
  const float* W_out = (const float*)d_in[4];
  float* out = (float*)d_out;

  char* ws = (char*)d_ws;
  float* P       = (float*)(ws);                          // 32768*512*4 = 64 MiB
  float* Wc      = (float*)(ws + (size_t)67108864);       // 512*512*4   =  1 MiB
  float* partial = (float*)(ws + (size_t)68157440);       // 8*32*512*4  = 512 KiB
  float* colsum  = (float*)(ws + (size_t)68681728);       // 8*512*4     =  16 KiB

  const int M_proj = BATCH * S_LEN;  // 32768

  // 1) Wc = W_out @ W_in
  gemm_wmma_f32<false><<<EMB / 32, 256, 0, stream>>>(W_out, W_in, Wc, EMB, EMB, EMB);

  // 2) P = values_flat @ Wc^T
  gemm_wmma_f32<true><<<M_proj / 32, 256, 0, stream>>>(values, Wc, P, M_proj, EMB, EMB);

  // 3) colsum over sequence axis (deterministic two-phase reduction)
  colsum_partial_kernel<<<dim3(32, BATCH), 128, 0, stream>>>(P, partial);
  colsum_final_kernel<<<BATCH, 128, 0, stream>>>(partial, colsum);

  // 4) banded softmax apply + normalize
  const size_t total4 = (size_t)BATCH * S_LEN * (EMB / 4);  // 4,194,304
  band_apply_kernel<<<(unsigned)(total4 / 256), 256, 0, stream>>>(P, colsum, out);
}